// DiffNet_86036784873567
// MI455X (gfx1250) — compile-verified
//
#include <hip/hip_runtime.h>
#include <math.h>

#define NF 512   // feature dim
#define RD 32    // reduce/embedding dim

typedef float v2f __attribute__((ext_vector_type(2)));
typedef float v8f __attribute__((ext_vector_type(8)));

// ---------------------------------------------------------------- zero
__global__ void zero_f32(float* __restrict__ p, long n) {
  long i  = (long)blockIdx.x * blockDim.x + threadIdx.x;
  long st = (long)gridDim.x * blockDim.x;
  for (; i < n; i += st) p[i] = 0.0f;
}

// ------------------------------------------------- global sum / sumsq (f64)
__global__ void stats_f32(const float4* __restrict__ x, long n4,
                          double* __restrict__ st2) {
  double s = 0.0, q = 0.0;
  long i  = (long)blockIdx.x * blockDim.x + threadIdx.x;
  long st = (long)gridDim.x * blockDim.x;
  for (; i < n4; i += st) {
    float4 v = x[i];
    s += (double)v.x + (double)v.y + (double)v.z + (double)v.w;
    q += (double)v.x * v.x + (double)v.y * v.y +
         (double)v.z * v.z + (double)v.w * v.w;
  }
  __shared__ double sh[512];
  int t = threadIdx.x;
  sh[t] = s; sh[256 + t] = q;
  __syncthreads();
  for (int k = 128; k > 0; k >>= 1) {
    if (t < k) { sh[t] += sh[t + k]; sh[256 + t] += sh[256 + t + k]; }
    __syncthreads();
  }
  if (t == 0) {
    atomicAdd(&st2[0], sh[0]);
    atomicAdd(&st2[1], sh[256]);
  }
}

// ------------------------------------------- mean / 1/std (ddof=1), 2 pairs
__global__ void finalize2(double* __restrict__ stats, float* __restrict__ scal,
                          double n0, double n1) {
  int t = threadIdx.x;           // 2 threads
  if (t > 1) return;
  double n    = t ? n1 : n0;
  double s    = stats[t * 2 + 0];
  double q    = stats[t * 2 + 1];
  double mean = s / n;
  double var  = (q - n * mean * mean) / (n - 1.0);
  scal[t * 2 + 0] = (float)mean;
  scal[t * 2 + 1] = (float)(1.0 / sqrt(var));
}

// ------------------------------------------------- column sums of W (512x32)
__global__ void wcs_kernel(const float* __restrict__ W, float* __restrict__ wcs) {
  int d = threadIdx.x;           // 32 threads
  float s = 0.0f;
  for (int k = 0; k < NF; ++k) s += W[k * RD + d];
  wcs[d] = s;
}

// ---------------------------------------------------------------- WMMA GEMM
// out[N x 32] = sigmoid( ((X - mu)/sigma) @ W + b )
//             = sigmoid( (X@W)*istd + (b - mu*istd*colsum(W)) )
// One wave computes one 16x16 tile via V_WMMA_F32_16X16X4_F32.
// blockDim = 64 (2 waves: col-half 0 and 1); gridDim = ceil(nrows/16).
__global__ void reduce_gemm(const float* __restrict__ A, int nrows,
                            const float* __restrict__ W,
                            const float* __restrict__ bias,
                            const float* __restrict__ wcs,
                            const float* __restrict__ scal,   // [mean, istd]
                            float* __restrict__ out,
                            double* __restrict__ st2) {
  const int tid  = threadIdx.x;
  const int lane = tid & 31;
  const int half = lane >> 4;    // 0: lanes 0-15, 1: lanes 16-31
  const int lm   = lane & 15;
  const int colh = tid >> 5;     // wave id -> which 16-col half of the 32 outs
  const int rb   = blockIdx.x;
  const int rbase = rb * 16;

  int arow = rbase + lm;
  if (arow > nrows - 1) arow = nrows - 1;            // uniform clamp, EXEC stays full
  const float* Ap = A + (long)arow * NF + half * 2;  // lane's 2 consecutive K elems
  const int col = colh * 16 + lm;

  v8f c = {};
#pragma unroll 4
  for (int k0 = 0; k0 < NF; k0 += 4) {
    // A operand (16x4 f32): v0 = K=k0 / K=k0+2 ; v1 = K=k0+1 / K=k0+3
    v2f a = *(const v2f*)(Ap + k0);
    // B operand (4x16 f32): mirrored striping, lane = column
    v2f b;
    b.x = W[(k0 + half * 2 + 0) * RD + col];
    b.y = W[(k0 + half * 2 + 1) * RD + col];
    c = __builtin_amdgcn_wmma_f32_16x16x4_f32(false, a, false, b,
                                              (short)0, c, false, false);
  }

  const float mean  = scal[0];
  const float istd  = scal[1];
  const float shift = bias[col] - mean * istd * wcs[col];

  double s = 0.0, q = 0.0;
#pragma unroll
  for (int j = 0; j < 8; ++j) {
    int row = rbase + half * 8 + j;   // C/D layout: vgpr j -> M=j (lanes<16) / M=8+j
    if (row < nrows) {
      float z  = c[j] * istd + shift;
      float sg = 1.0f / (1.0f + expf(-z));
      out[(long)row * RD + col] = sg;
      s += (double)sg;
      q += (double)sg * sg;
    }
  }

  __shared__ double sh[128];
  sh[tid] = s; sh[64 + tid] = q;
  __syncthreads();
  for (int k = 32; k > 0; k >>= 1) {
    if (tid < k) { sh[tid] += sh[tid + k]; sh[64 + tid] += sh[64 + tid + k]; }
    __syncthreads();
  }
  if (tid == 0) {
    atomicAdd(&st2[0], sh[0]);
    atomicAdd(&st2[1], sh[64]);
  }
}

// ----------------------------------------- fusion: x = (x-mu)*istd + emb
__global__ void fusion_kernel(float* __restrict__ red,
                              const float* __restrict__ emb, long n,
                              const float* __restrict__ scal) {
  const float mean = scal[0];
  const float istd = scal[1];
  long i  = (long)blockIdx.x * blockDim.x + threadIdx.x;
  long st = (long)gridDim.x * blockDim.x;
  for (; i < n; i += st)
    red[i] = (red[i] - mean) * istd + emb[i];
}

// --------------------------------- COO SpMM: out[r] += v * X[c]  (wave/edge)
__global__ void spmm_atomic(const int* __restrict__ rows,
                            const int* __restrict__ cols,
                            const float* __restrict__ vals,
                            const float* __restrict__ X,
                            float* __restrict__ out, int nnz) {
  long t = (long)blockIdx.x * blockDim.x + threadIdx.x;
  int e = (int)(t >> 5);
  int d = (int)(t & 31);
  if (e < nnz) {
    int   r = rows[e];
    int   c = cols[e];
    float v = vals[e];
    atomicAdd(&out[(long)r * RD + d], v * X[(long)c * RD + d]);
  }
}

// ------------------------- gather + dot(32) + sigmoid, one wave per pair
__global__ void predict_kernel(const float* __restrict__ LU,
                               const float* __restrict__ IF,
                               const int* __restrict__ upi,
                               const int* __restrict__ ipi,
                               float* __restrict__ out, int npairs) {
  long t = (long)blockIdx.x * blockDim.x + threadIdx.x;
  int p = (int)(t >> 5);
  int d = (int)(t & 31);
  if (p >= npairs) return;
  int u  = upi[p];
  int it = ipi[p];
  float v = LU[(long)u * RD + d] * IF[(long)it * RD + d];
#pragma unroll
  for (int off = 16; off > 0; off >>= 1)
    v += __shfl_xor(v, off, 32);
  if (d == 0) {
    out[p] = v;
    out[(long)npairs + p] = 1.0f / (1.0f + expf(-v));
  }
}

// ================================================================ launch
extern "C" void kernel_launch(void* const* d_in, const int* in_sizes, int n_in,
                              void* d_out, int out_size, void* d_ws, size_t ws_size,
                              hipStream_t stream) {
  const float* uf   = (const float*)d_in[0];
  const float* itf  = (const float*)d_in[1];
  const float* uemb = (const float*)d_in[2];
  const float* iemb = (const float*)d_in[3];
  const float* W    = (const float*)d_in[4];
  const float* bia  = (const float*)d_in[5];
  const int*   s_r  = (const int*)d_in[6];
  const int*   s_c  = (const int*)d_in[7];
  const float* s_v  = (const float*)d_in[8];
  const int*   ui_r = (const int*)d_in[9];
  const int*   ui_c = (const int*)d_in[10];
  const float* ui_v = (const float*)d_in[11];
  const int*   p_u  = (const int*)d_in[12];
  const int*   p_i  = (const int*)d_in[13];
  float* out = (float*)d_out;

  const int  nU   = in_sizes[2] / RD;    // 100000
  const int  nI   = in_sizes[3] / RD;    // 50000
  const long nUF  = (long)in_sizes[0];   // 51.2M
  const long nIF  = (long)in_sizes[1];   // 25.6M
  const int  eSoc = in_sizes[6];
  const int  eUI  = in_sizes[9];
  const int  nP   = in_sizes[12];

  // workspace layout
  char*   ws    = (char*)d_ws;
  double* stats = (double*)ws;             // 8 doubles: uf,itf,usig,isig pairs
  float*  scal  = (float*)(ws + 64);       // 8 floats:  mean/istd pairs
  float*  wcs   = (float*)(ws + 96);       // 32 floats: colsum(W)
  float*  uRed  = (float*)(ws + 256);                  // 100000x32 (-> user_fusion)
  float*  iRed  = uRed + (long)nU * RD;                // 50000x32  (-> item_fusion)
  float*  Lbuf  = iRed + (long)nI * RD;                // item_consumed + h2
  float*  H1    = Lbuf + (long)nU * RD;                // h1

  // init accumulators
  zero_f32<<<1, 64, 0, stream>>>((float*)ws, 56);                      // stats region
  zero_f32<<<2048, 256, 0, stream>>>(Lbuf, (long)nU * RD * 2);         // L and H1

  // stage-1 stats: raw feature mean/std
  stats_f32<<<2048, 256, 0, stream>>>((const float4*)uf,  nUF / 4, stats + 0);
  stats_f32<<<2048, 256, 0, stream>>>((const float4*)itf, nIF / 4, stats + 2);
  wcs_kernel<<<1, 32, 0, stream>>>(W, wcs);
  finalize2<<<1, 2, 0, stream>>>(stats, scal, (double)nUF, (double)nIF);

  // fused (normalize @ W + b -> sigmoid) with WMMA; accumulate stage-2 stats
  reduce_gemm<<<(nU + 15) / 16, 64, 0, stream>>>(uf,  nU, W, bia, wcs, scal + 0, uRed, stats + 4);
  reduce_gemm<<<(nI + 15) / 16, 64, 0, stream>>>(itf, nI, W, bia, wcs, scal + 2, iRed, stats + 6);
  finalize2<<<1, 2, 0, stream>>>(stats + 4, scal + 4,
                                 (double)((long)nU * RD), (double)((long)nI * RD));

  // fusion = normalize(reduce) + embedding (in place)
  fusion_kernel<<<2048, 256, 0, stream>>>(uRed, uemb, (long)nU * RD, scal + 4);
  fusion_kernel<<<1024, 256, 0, stream>>>(iRed, iemb, (long)nI * RD, scal + 6);

  // SpMMs (wave per edge, lane per dim)
  long tUI = (long)eUI * 32;
  spmm_atomic<<<(unsigned)((tUI + 255) / 256), 256, 0, stream>>>(ui_r, ui_c, ui_v, iRed, Lbuf, eUI);
  long tS = (long)eSoc * 32;
  spmm_atomic<<<(unsigned)((tS + 255) / 256), 256, 0, stream>>>(s_r, s_c, s_v, uRed, H1, eSoc);
  spmm_atomic<<<(unsigned)((tS + 255) / 256), 256, 0, stream>>>(s_r, s_c, s_v, H1, Lbuf, eSoc);

  // final gather-dot-sigmoid
  long tP = (long)nP * 32;
  predict_kernel<<<(unsigned)((tP + 255) / 256), 256, 0, stream>>>(Lbuf, iRed, p_u, p_i, out, nP);
}